// TrajectorySSM_59614146068526
// MI455X (gfx1250) — compile-verified
//
#include <hip/hip_runtime.h>
#include <hip/hip_bf16.h>

// ---------------------------------------------------------------------------
// TrajectorySSM on MI455X (gfx1250), fp32 WMMA (V_WMMA_F32_16X16X4_F32).
//
//   1) proj_in : Bx[s,b,n]   = sum_d x[b,s,d] * B[n,d]     (WMMA, LDS-staged W)
//   2) matsq x5: AL = A^32 (chunk carry propagation)
//   3) scan_local : per-chunk local scan (zero init), 64 chunks x 32 steps
//   4) scan_carry : sequential carry over 64 chunks w/ AL, emits h_final
//   5) scan_fix   : hs[s] += H_chunk * (A^T)^(t+1)  (parallel correction)
//   6) proj_out : y = hs@C^T + x@D^T                (WMMA, LDS-staged W)
//
// Weight tiles are staged global->LDS with async-to-LDS copies (ASYNCcnt)
// when the toolchain exposes the gfx1250 builtin; otherwise a synchronous
// b128 copy fallback keeps the code compiling identically.
// ---------------------------------------------------------------------------

typedef __attribute__((ext_vector_type(2))) float v2f;
typedef __attribute__((ext_vector_type(4))) float v4f;
typedef __attribute__((ext_vector_type(8))) float v8f;
typedef __attribute__((ext_vector_type(4))) int   v4i;

constexpr int BATCH  = 8;
constexpr int SEQ    = 2048;
constexpr int DMODEL = 1024;
constexpr int STATE  = 256;
constexpr int CHUNK  = 32;           // chunk length
constexpr int NCHUNK = 64;           // SEQ / CHUNK
constexpr int LDSPITCH = STATE + 2;  // scan-state pitch (bank-conflict pad)
constexpr int KT     = 64;           // K-tile staged in LDS for proj GEMMs
constexpr int WPITCH = 68;           // weight-tile pitch: 16B aligned, banks ok

// D = A(16x4 f32) * B(4x16 f32) + C(16x16 f32)
__device__ __forceinline__ v8f wmma4(v2f a, v2f b, v8f c) {
  return __builtin_amdgcn_wmma_f32_16x16x4_f32(
      /*neg_a=*/false, a, /*neg_b=*/false, b,
      /*c_mod=*/(short)0, c, /*reuse_a=*/false, /*reuse_b=*/false);
}

// --------------------------- async LDS staging -----------------------------
#if __has_builtin(__builtin_amdgcn_global_load_async_to_lds_b128)
#define SSM_ASYNC_LDS 1
#endif

__device__ __forceinline__ void async_drain() {
#if defined(SSM_ASYNC_LDS) && __has_builtin(__builtin_amdgcn_s_wait_asynccnt)
  __builtin_amdgcn_s_wait_asynccnt(0);
#endif
}

// Copy a 64(N) x 64(K) f32 weight tile into lds[64][WPITCH].
__device__ __forceinline__ void stage_W(const float* __restrict__ W, int ldw,
                                        int nbase, int k0,
                                        float* __restrict__ lds) {
  const int seg = threadIdx.x & 15;   // 16B segment within a 64-float row
  const int r0  = threadIdx.x >> 4;   // 0..15
#pragma unroll
  for (int rr = 0; rr < 64; rr += 16) {
    const int row = r0 + rr;
    const float* g = W + (size_t)(nbase + row) * ldw + k0 + seg * 4;
    float* l = lds + row * WPITCH + seg * 4;
#if defined(SSM_ASYNC_LDS)
    __builtin_amdgcn_global_load_async_to_lds_b128((v4i*)g, (v4i*)l, 0, 0);
#else
    *(v4f*)l = *(const v4f*)g;
#endif
  }
}

// ---------------------------------------------------------------------------
// Wave-level GEMM: one wave computes a 32(M) x 64(N) tile, Y = X * W^T.
// X rows come from per-lane row pointers (lane%16 selects M within strip);
// W tiles are double-buffered in LDS.  A-frag: lane L -> M=L%16, K pair
// (L/16)*2.  B-frag: lane L -> N = nloc(+16t), same K pair.
// ---------------------------------------------------------------------------
__device__ __forceinline__ void gemm_block(const float* __restrict__ xrow0,
                                           const float* __restrict__ xrow1,
                                           const float* __restrict__ W,
                                           int ldw, int nbase, int K,
                                           float* __restrict__ Wl,  // 2 bufs
                                           v8f acc[8]) {
  const int lane = threadIdx.x & 31;
  const int kb   = ((lane >> 4) & 1) << 1;
  const int nloc = lane & 15;
  int buf = 0;
  stage_W(W, ldw, nbase, 0, Wl);
  for (int k0 = 0; k0 < K; k0 += KT) {
    async_drain();
    __syncthreads();  // staged tile visible; prev tile no longer being read
    if (k0 + KT < K) stage_W(W, ldw, nbase, k0 + KT, Wl + (buf ^ 1) * (64 * WPITCH));
    const float* Wb = Wl + buf * (64 * WPITCH);
    for (int kk = 0; kk < KT; kk += 4) {
      v2f a0 = *(const v2f*)(xrow0 + k0 + kk + kb);
      v2f a1 = *(const v2f*)(xrow1 + k0 + kk + kb);
#pragma unroll
      for (int t = 0; t < 4; ++t) {
        v2f b = *(const v2f*)&Wb[(nloc + 16 * t) * WPITCH + kk + kb];
        acc[t]     = wmma4(a0, b, acc[t]);
        acc[4 + t] = wmma4(a1, b, acc[4 + t]);
      }
    }
    __syncthreads();  // everyone done reading this buffer
    buf ^= 1;
  }
}

// ---------------------------------------------------------------------------
// Kernel 1: Bx[s*8+b, n] = sum_d x[b,s,d] * B[n,d]  (M=16384, N=256, K=1024)
// ---------------------------------------------------------------------------
__global__ void __launch_bounds__(256) proj_in_kernel(
    const float* __restrict__ x, const float* __restrict__ B,
    float* __restrict__ Bx) {
  __shared__ float Wl[2 * 64 * WPITCH];
  const int lane  = threadIdx.x & 31;
  const int wave  = threadIdx.x >> 5;
  const int mw    = blockIdx.y * 256 + wave * 32;
  const int nbase = blockIdx.x * 64;
  const int m0 = mw + (lane & 15);        // output rows are s*8+b ordered
  const int m1 = m0 + 16;
  const float* xrow0 = x + ((size_t)(m0 & 7) * SEQ + (m0 >> 3)) * DMODEL;
  const float* xrow1 = x + ((size_t)(m1 & 7) * SEQ + (m1 >> 3)) * DMODEL;
  __builtin_prefetch(xrow0, 0, 0);
  v8f acc[8] = {{}, {}, {}, {}, {}, {}, {}, {}};
  gemm_block(xrow0, xrow1, B, DMODEL, nbase, DMODEL, Wl, acc);
  const int mo   = mw + ((lane >> 4) << 3);
  const int nloc = lane & 15;
#pragma unroll
  for (int t = 0; t < 4; ++t)
#pragma unroll
    for (int r = 0; r < 8; ++r) {
      Bx[(size_t)(mo + r) * STATE + nbase + t * 16 + nloc]      = acc[t][r];
      Bx[(size_t)(mo + 16 + r) * STATE + nbase + t * 16 + nloc] = acc[4 + t][r];
    }
}

// ---------------------------------------------------------------------------
// Kernel 2 (x5): Dst = S @ S  (256x256 fp32 squaring for A^32)
// ---------------------------------------------------------------------------
__global__ void __launch_bounds__(256) matsq_kernel(
    const float* __restrict__ S, float* __restrict__ Dst) {
  const int i = blockIdx.x, j = threadIdx.x;
  float sum = 0.f;
  for (int k = 0; k < STATE; ++k)
    sum = fmaf(S[(size_t)i * STATE + k], S[(size_t)k * STATE + j], sum);
  Dst[(size_t)i * STATE + j] = sum;
}

// ---------------------------------------------------------------------------
// Scan helpers: H (16x256, batch padded to 16) in LDS; one step computes
// H @ W^T with W row-major (256,256).  8 waves, each owns 32 columns.
// ---------------------------------------------------------------------------
__device__ __forceinline__ void hstep(const float* __restrict__ W,
                                      const float* __restrict__ Hlds,
                                      v8f acc[2]) {
  const int lane = threadIdx.x & 31;
  const int wave = threadIdx.x >> 5;
  const int mrow = lane & 15;
  const int kb   = ((lane >> 4) & 1) << 1;
  const int n0   = wave * 32 + (lane & 15);
  acc[0] = (v8f){};
  acc[1] = (v8f){};
  for (int k0 = 0; k0 < STATE; k0 += 4) {
    v2f a  = *(const v2f*)&Hlds[mrow * LDSPITCH + k0 + kb];
    v2f b0 = *(const v2f*)&W[(size_t)n0 * STATE + k0 + kb];
    v2f b1 = *(const v2f*)&W[(size_t)(n0 + 16) * STATE + k0 + kb];
    acc[0] = wmma4(a, b0, acc[0]);
    acc[1] = wmma4(a, b1, acc[1]);
  }
}

__device__ __forceinline__ void hwrite(float* __restrict__ Hlds,
                                       const v8f acc[2]) {
  const int lane = threadIdx.x & 31;
  const int wave = threadIdx.x >> 5;
  const int mo   = (lane >> 4) << 3;  // C/D layout: M = r + 8*(lane/16)
  const int col  = wave * 32 + (lane & 15);
#pragma unroll
  for (int r = 0; r < 8; ++r) {
    Hlds[(mo + r) * LDSPITCH + col]      = acc[0][r];
    Hlds[(mo + r) * LDSPITCH + col + 16] = acc[1][r];
  }
}

// ---------------------------------------------------------------------------
// Kernel 3: local scan per chunk (h starts at 0).  In-place on Bx buffer;
// records chunk-end state E[c].
// ---------------------------------------------------------------------------
__global__ void __launch_bounds__(256) scan_local_kernel(
    const float* __restrict__ A, float* __restrict__ hsBx,
    float* __restrict__ E) {
  __shared__ float Hlds[16 * LDSPITCH];
  const int c = blockIdx.x;
  for (int i = threadIdx.x; i < 16 * LDSPITCH; i += blockDim.x) Hlds[i] = 0.f;
  __syncthreads();
  const int lane = threadIdx.x & 31;
  const int wave = threadIdx.x >> 5;
  const int col  = wave * 32 + (lane & 15);
  v8f acc[2];
  for (int t = 0; t < CHUNK; ++t) {
    const int s = c * CHUNK + t;
    hstep(A, Hlds, acc);
    if (lane < 16) {  // real batch rows live in lanes 0..15 (M = r < 8)
#pragma unroll
      for (int r = 0; r < 8; ++r) {
        const size_t base = ((size_t)s * BATCH + r) * STATE;
        acc[0][r] += hsBx[base + col];
        acc[1][r] += hsBx[base + col + 16];
      }
    }
    __syncthreads();  // all waves done reading H
    hwrite(Hlds, acc);
    if (lane < 16) {
#pragma unroll
      for (int r = 0; r < 8; ++r) {
        const size_t base = ((size_t)s * BATCH + r) * STATE;
        hsBx[base + col]      = acc[0][r];
        hsBx[base + col + 16] = acc[1][r];
      }
    }
    __syncthreads();  // H fully written before next step
  }
  if (lane < 16) {
#pragma unroll
    for (int r = 0; r < 8; ++r) {
      const size_t base = ((size_t)c * BATCH + r) * STATE;
      E[base + col]      = acc[0][r];
      E[base + col + 16] = acc[1][r];
    }
  }
}

// ---------------------------------------------------------------------------
// Kernel 4: sequential carry across chunks: H_c = H_{c-1}*(A^T)^32 + E[c-1].
// Single workgroup, NCHUNK steps.  Emits Hc[c] and final state.
// ---------------------------------------------------------------------------
__global__ void __launch_bounds__(256) scan_carry_kernel(
    const float* __restrict__ AL, const float* __restrict__ h0,
    const float* __restrict__ E, float* __restrict__ Hc,
    float* __restrict__ hfinal) {
  __shared__ float Hlds[16 * LDSPITCH];
  for (int i = threadIdx.x; i < 16 * LDSPITCH; i += blockDim.x) {
    const int row = i / LDSPITCH, k = i - row * LDSPITCH;
    Hlds[i] = (row < BATCH && k < STATE) ? h0[k] : 0.f;
  }
  for (int j = threadIdx.x; j < BATCH * STATE; j += blockDim.x)
    Hc[j] = h0[j & (STATE - 1)];  // Hc[0] = broadcast h0
  __syncthreads();
  const int lane = threadIdx.x & 31;
  const int wave = threadIdx.x >> 5;
  const int col  = wave * 32 + (lane & 15);
  v8f acc[2];
  for (int c = 1; c <= NCHUNK; ++c) {
    hstep(AL, Hlds, acc);
    if (lane < 16) {
#pragma unroll
      for (int r = 0; r < 8; ++r) {
        const size_t base = ((size_t)(c - 1) * BATCH + r) * STATE;
        acc[0][r] += E[base + col];
        acc[1][r] += E[base + col + 16];
      }
    }
    __syncthreads();
    hwrite(Hlds, acc);
    if (lane < 16) {
      if (c < NCHUNK) {
#pragma unroll
        for (int r = 0; r < 8; ++r) {
          const size_t base = ((size_t)c * BATCH + r) * STATE;
          Hc[base + col]      = acc[0][r];
          Hc[base + col + 16] = acc[1][r];
        }
      } else {
#pragma unroll
        for (int r = 0; r < 8; ++r) {
          hfinal[(size_t)r * STATE + col]      = acc[0][r];
          hfinal[(size_t)r * STATE + col + 16] = acc[1][r];
        }
      }
    }
    __syncthreads();
  }
}

// ---------------------------------------------------------------------------
// Kernel 5: correction: hs[c*32+t] += Hc[c] * (A^T)^(t+1), parallel per chunk.
// ---------------------------------------------------------------------------
__global__ void __launch_bounds__(256) scan_fix_kernel(
    const float* __restrict__ A, const float* __restrict__ Hc,
    float* __restrict__ hs) {
  __shared__ float Hlds[16 * LDSPITCH];
  const int c = blockIdx.x;
  for (int i = threadIdx.x; i < 16 * LDSPITCH; i += blockDim.x) {
    const int row = i / LDSPITCH, k = i - row * LDSPITCH;
    Hlds[i] = (row < BATCH && k < STATE)
                  ? Hc[((size_t)c * BATCH + row) * STATE + k] : 0.f;
  }
  __syncthreads();
  const int lane = threadIdx.x & 31;
  const int wave = threadIdx.x >> 5;
  const int col  = wave * 32 + (lane & 15);
  v8f acc[2];
  for (int t = 0; t < CHUNK; ++t) {
    const int s = c * CHUNK + t;
    hstep(A, Hlds, acc);
    __syncthreads();
    hwrite(Hlds, acc);
    if (lane < 16) {
#pragma unroll
      for (int r = 0; r < 8; ++r) {
        const size_t base = ((size_t)s * BATCH + r) * STATE;
        hs[base + col]      += acc[0][r];
        hs[base + col + 16] += acc[1][r];
      }
    }
    __syncthreads();
  }
}

// ---------------------------------------------------------------------------
// Kernel 6: y[m,d] = hs_row(m) @ C^T + x_row(m) @ D^T   (m = b*SEQ + s)
// ---------------------------------------------------------------------------
__global__ void __launch_bounds__(256) proj_out_kernel(
    const float* __restrict__ x, const float* __restrict__ hs,
    const float* __restrict__ C, const float* __restrict__ D,
    float* __restrict__ y) {
  __shared__ float Wl[2 * 64 * WPITCH];
  const int lane  = threadIdx.x & 31;
  const int wave  = threadIdx.x >> 5;
  const int mw    = blockIdx.y * 256 + wave * 32;
  const int nbase = blockIdx.x * 64;
  const int m0 = mw + (lane & 15);
  const int m1 = m0 + 16;
  const int b0 = m0 >> 11, s0 = m0 & (SEQ - 1);
  const int b1 = m1 >> 11, s1 = m1 & (SEQ - 1);
  const float* hrow0 = hs + ((size_t)s0 * BATCH + b0) * STATE;
  const float* hrow1 = hs + ((size_t)s1 * BATCH + b1) * STATE;
  const float* xrow0 = x + (size_t)m0 * DMODEL;
  const float* xrow1 = x + (size_t)m1 * DMODEL;
  __builtin_prefetch(xrow0, 0, 0);
  v8f acc[8] = {{}, {}, {}, {}, {}, {}, {}, {}};
  gemm_block(hrow0, hrow1, C, STATE, nbase, STATE, Wl, acc);
  gemm_block(xrow0, xrow1, D, DMODEL, nbase, DMODEL, Wl, acc);
  const int mo   = mw + ((lane >> 4) << 3);
  const int nloc = lane & 15;
#pragma unroll
  for (int t = 0; t < 4; ++t)
#pragma unroll
    for (int r = 0; r < 8; ++r) {
      y[(size_t)(mo + r) * DMODEL + nbase + t * 16 + nloc]      = acc[t][r];
      y[(size_t)(mo + 16 + r) * DMODEL + nbase + t * 16 + nloc] = acc[4 + t][r];
    }
}

// ---------------------------------------------------------------------------
extern "C" void kernel_launch(void* const* d_in, const int* in_sizes, int n_in,
                              void* d_out, int out_size, void* d_ws,
                              size_t ws_size, hipStream_t stream) {
  const float* x  = (const float*)d_in[0];
  const float* A  = (const float*)d_in[1];
  const float* B  = (const float*)d_in[2];
  const float* C  = (const float*)d_in[3];
  const float* D  = (const float*)d_in[4];
  const float* h0 = (const float*)d_in[5];

  float* y      = (float*)d_out;
  float* hfinal = y + (size_t)BATCH * SEQ * DMODEL;

  float* ws   = (float*)d_ws;
  float* hsBx = ws;                                  // 4,194,304 f32
  float* E    = hsBx + (size_t)SEQ * BATCH * STATE;  //   131,072 f32
  float* Hc   = E + (size_t)NCHUNK * BATCH * STATE;  //   131,072 f32
  float* Pa   = Hc + (size_t)NCHUNK * BATCH * STATE; //    65,536 f32
  float* Pb   = Pa + (size_t)STATE * STATE;          //    65,536 f32

  // 1) input projection
  proj_in_kernel<<<dim3(STATE / 64, (BATCH * SEQ) / 256), 256, 0, stream>>>(
      x, B, hsBx);

  // 2) A^32 via repeated squaring (A^2, A^4, A^8, A^16, A^32)
  matsq_kernel<<<STATE, STATE, 0, stream>>>(A, Pa);
  matsq_kernel<<<STATE, STATE, 0, stream>>>(Pa, Pb);
  matsq_kernel<<<STATE, STATE, 0, stream>>>(Pb, Pa);
  matsq_kernel<<<STATE, STATE, 0, stream>>>(Pa, Pb);
  matsq_kernel<<<STATE, STATE, 0, stream>>>(Pb, Pa);  // Pa = A^32

  // 3) parallel local scans
  scan_local_kernel<<<NCHUNK, 256, 0, stream>>>(A, hsBx, E);
  // 4) sequential carry across chunks
  scan_carry_kernel<<<1, 256, 0, stream>>>(Pa, h0, E, Hc, hfinal);
  // 5) parallel corrections
  scan_fix_kernel<<<NCHUNK, 256, 0, stream>>>(A, Hc, hsBx);

  // 6) output projection
  proj_out_kernel<<<dim3(DMODEL / 64, (BATCH * SEQ) / 256), 256, 0, stream>>>(
      x, hsBx, C, D, y);
}